// QuantumTemporalReservoir_67894843015548
// MI455X (gfx1250) — compile-verified
//
#include <hip/hip_runtime.h>
#include <cmath>

#define PI_F 3.14159265358979323846f

typedef float v2f __attribute__((ext_vector_type(2)));
typedef float v8f __attribute__((ext_vector_type(8)));

// ---------------------------------------------------------------------------
// Kernel A: angles[B,16] = tanh(x @ W^T) * pi  using V_WMMA_F32_16X16X4_F32.
// 256 threads = 8 waves per block; each wave owns one 16-row batch tile.
// Per lane the two fragment elements are CONSECUTIVE floats (k0=kk+2*half,
// k1=k0+1) -> one global_load_b64 each for A and B per K-step. Padded B
// columns (12..15) are left unmasked: their output columns are never read.
// ---------------------------------------------------------------------------
__global__ __launch_bounds__(256) void qtr_angles_wmma(
    const float* __restrict__ x,      // [B,64]
    const float* __restrict__ W,      // [12,64]
    float* __restrict__ angles)       // [B,16] (cols 12..15 garbage, unused)
{
    const int lane = threadIdx.x & 31;
    const int wave = threadIdx.x >> 5;
    const int half = lane >> 4;       // 0 or 1
    const int col  = lane & 15;       // M for A-frag, N for B/D-frags
    const int m0   = (blockIdx.x * 8 + wave) * 16;
    const int wrow = (col < 12) ? col : 0;   // clamp; padded cols unread

    const float* __restrict__ xp = x + (size_t)(m0 + col) * 64 + 2 * half;
    const float* __restrict__ wp = W + (size_t)wrow * 64 + 2 * half;

    v8f acc = {};
    #pragma unroll
    for (int kk = 0; kk < 64; kk += 4) {
        const v2f a = *(const v2f*)(xp + kk);   // global_load_b64
        const v2f b = *(const v2f*)(wp + kk);   // global_load_b64
        acc = __builtin_amdgcn_wmma_f32_16x16x4_f32(
            /*neg_a=*/false, a, /*neg_b=*/false, b,
            /*c_mod=*/(short)0, acc, /*reuse_a=*/false, /*reuse_b=*/false);
    }

    #pragma unroll
    for (int r = 0; r < 8; ++r) {
        const int m = m0 + r + 8 * half;
        angles[m * 16 + col] = tanhf(acc[r]) * PI_F;
    }
}

// ---------------------------------------------------------------------------
// Kernel B: 12-qubit state-vector simulation, one batch item per workgroup.
// State (4096 complex) lives entirely in LDS (32 KB) as interleaved float2
// (re,im) -> every amplitude access is a single 64-bit DS op (ds_load_b64 /
// ds_store_b64), halving DS instruction count vs split re/im arrays, and
// 32 consecutive lanes span exactly the 64 LDS banks (conflict-free).
// Per layer per qubit, RY(data)*RY(theta)*RZ(phi) fused into one 2x2
// unitary: U = diag(e^{-i phi/2}, e^{+i phi/2}) * RY(alpha + theta).
// CNOT = conditional swap (pure LDS permutation).
// ---------------------------------------------------------------------------
__global__ __launch_bounds__(256) void qtr_sim(
    const float* __restrict__ angles, // [B,16]
    const float* __restrict__ qp,     // [3,12,2]
    float* __restrict__ out)          // [B,12]
{
    __shared__ float2 sA[4096];       // interleaved (re, im), 32 KB
    __shared__ float  red[16];

    const int tid = threadIdx.x;
    const int b   = blockIdx.x;

    if (tid < 16) red[tid] = 0.0f;
    #pragma unroll
    for (int j = 0; j < 16; ++j)
        sA[tid + 256 * j] = make_float2(0.0f, 0.0f);
    if (tid == 0) sA[0] = make_float2(1.0f, 0.0f);   // |0...0>
    __syncthreads();

    for (int layer = 0; layer < 3; ++layer) {
        // ---- fused single-qubit gates ----
        for (int q = 0; q < 12; ++q) {
            const float theta = qp[(layer * 12 + q) * 2 + 0];
            const float phi   = qp[(layer * 12 + q) * 2 + 1];
            const float alpha = angles[b * 16 + q];
            const float beta  = 0.5f * (alpha + theta);
            const float c  = cosf(beta);
            const float s  = sinf(beta);
            const float hp = 0.5f * phi;
            const float pr  =  cosf(hp);   // e^{-i phi/2} = pr + i*pim
            const float pim = -sinf(hp);

            const int sh     = 11 - q;       // wire 0 = most significant
            const int stride = 1 << sh;
            const int low    = stride - 1;

            #pragma unroll
            for (int j = 0; j < 8; ++j) {
                const int p  = tid + 256 * j;                    // 0..2047
                const int i0 = ((p & ~low) << 1) | (p & low);    // bit_q = 0
                const int i1 = i0 + stride;                      // bit_q = 1
                const float2 a0 = sA[i0];                        // ds_load_b64
                const float2 a1 = sA[i1];
                // RY part
                const float t0r = c * a0.x - s * a1.x, t0i = c * a0.y - s * a1.y;
                const float t1r = s * a0.x + c * a1.x, t1i = s * a0.y + c * a1.y;
                // RZ part: a0 *= (pr + i pim), a1 *= (pr - i pim)
                sA[i0] = make_float2(pr * t0r - pim * t0i,
                                     pr * t0i + pim * t0r);      // ds_store_b64
                sA[i1] = make_float2(pr * t1r + pim * t1i,
                                     pr * t1i - pim * t1r);
            }
            __syncthreads();
        }

        // ---- CNOT ring (conditional swaps) ----
        for (int q = 0; q < 12; ++q) {
            const int ctrl = q, tgt = (q + 1) % 12;
            const int pc = 11 - ctrl, pt = 11 - tgt;
            const int pl = pc < pt ? pc : pt;
            const int ph = pc < pt ? pt : pc;

            #pragma unroll
            for (int j = 0; j < 4; ++j) {
                const int w  = tid + 256 * j;                    // 0..1023
                // insert zero-bits at positions pl then ph (final index space)
                const int x1 = ((w >> pl) << (pl + 1)) | (w & ((1 << pl) - 1));
                int i        = ((x1 >> ph) << (ph + 1)) | (x1 & ((1 << ph) - 1));
                i |= (1 << pc);                                  // control = 1
                const int ip = i | (1 << pt);                    // target partner
                const float2 t = sA[i];
                sA[i]  = sA[ip];
                sA[ip] = t;
            }
            __syncthreads();
        }
    }

    // ---- <Z_q> = sum_i |amp_i|^2 * (1 - 2*bit_q(i)) ----
    float acc[12];
    #pragma unroll
    for (int q = 0; q < 12; ++q) acc[q] = 0.0f;

    #pragma unroll
    for (int j = 0; j < 16; ++j) {
        const int i   = tid + 256 * j;
        const float2 a = sA[i];
        const float p  = a.x * a.x + a.y * a.y;
        #pragma unroll
        for (int q = 0; q < 12; ++q)
            acc[q] += ((i >> (11 - q)) & 1) ? -p : p;
    }

    #pragma unroll
    for (int q = 0; q < 12; ++q)
        atomicAdd(&red[q], acc[q]);      // ds_add_f32
    __syncthreads();

    if (tid < 12) out[b * 12 + tid] = red[tid];
}

// ---------------------------------------------------------------------------
extern "C" void kernel_launch(void* const* d_in, const int* in_sizes, int n_in,
                              void* d_out, int out_size, void* d_ws, size_t ws_size,
                              hipStream_t stream) {
    (void)n_in; (void)out_size; (void)ws_size;
    const float* x  = (const float*)d_in[0];   // [8192,64]
    const float* W  = (const float*)d_in[1];   // [12,64]
    const float* qp = (const float*)d_in[2];   // [3,12,2]
    float* out      = (float*)d_out;           // [8192,12]
    float* angles   = (float*)d_ws;            // [8192,16] staging (512 KB)

    const int B = in_sizes[0] / 64;            // 8192

    qtr_angles_wmma<<<B / (16 * 8), 256, 0, stream>>>(x, W, angles);
    qtr_sim<<<B, 256, 0, stream>>>(angles, qp, out);
}